// ConvKernel_63806034149612
// MI455X (gfx1250) — compile-verified
//
#include <hip/hip_runtime.h>
#include <hip/hip_bf16.h>
#include <math.h>

typedef float v2f __attribute__((ext_vector_type(2)));
typedef float v8f __attribute__((ext_vector_type(8)));

#define DIM 2048
#define BATCH 8192
#define INDIM 512
#define BM 64
#define BN 128
#define BK 16
#define APAD 20   // As row stride: keeps 16B align, conflict-free frag reads
#define BPAD 18   // Bs row stride: even (8B align for b64), conflict-spread

// ---------------------------------------------------------------------------
// Dense fp32 GEMM, C = A(MxK) * B(KxN), row-major, V_WMMA_F32_16X16X4_F32.
// 256 threads = 8 waves; block tile 64x128; wave tile 16x64 (4 accumulators).
// LDS double-buffered; B stored transposed in LDS so every fragment is one
// aligned ds_load_b64 (no VGPR packing moves before the WMMA).
// ---------------------------------------------------------------------------
__global__ __launch_bounds__(256) void gemm_nn_wmma(
    const float* __restrict__ A, const float* __restrict__ B,
    float* __restrict__ C, int M, int N, int K) {
  __shared__ float As[2][BM][APAD];
  __shared__ float Bs[2][BN][BPAD];   // transposed: Bs[n][k]

  const int tid  = threadIdx.x;
  const int wave = tid >> 5;
  const int lane = tid & 31;
  const int bm = blockIdx.y * BM;
  const int bn = blockIdx.x * BN;

  const int tm = (wave & 3) * 16;      // wave's row tile within block
  const int tn = (wave >> 2) * 64;     // wave's 64-wide col strip (4 tiles)

  v8f acc0 = {}, acc1 = {}, acc2 = {}, acc3 = {};

  // cooperative tile-load coords
  const int arow = tid >> 2;           // 0..63
  const int acol = (tid & 3) * 4;      // 0,4,8,12
  const int brow = tid >> 4;           // 0..15   (k within tile)
  const int bcol = (tid & 15) * 8;     // 0..120  (8 consecutive n)

  // WMMA f32 fragment coords (ISA 7.12.2): lane&15 = M/N index, (lane>>4)*2 = K pair
  const int mfrag = lane & 15;
  const int kfrag = (lane >> 4) * 2;

  const float* ap = A + (size_t)(bm + arow) * K + acol;
  const float* bp = B + (size_t)brow * N + bn + bcol;

  // prologue: stage tile 0
  float4 av  = *(const float4*)ap;
  float4 bv0 = *(const float4*)bp;
  float4 bv1 = *(const float4*)(bp + 4);
  *(float4*)&As[0][arow][acol] = av;
#pragma unroll
  for (int i = 0; i < 4; ++i) {
    Bs[0][bcol + i][brow]     = ((const float*)&bv0)[i];
    Bs[0][bcol + 4 + i][brow] = ((const float*)&bv1)[i];
  }
  __syncthreads();

  int buf = 0;
  for (int k0 = 0; k0 < K; k0 += BK) {
    const bool more = (k0 + BK < K);
    if (more) {                         // fetch next tile while computing this one
      const float* apn = ap + (k0 + BK);
      const float* bpn = bp + (size_t)(k0 + BK) * N;
      av  = *(const float4*)apn;
      bv0 = *(const float4*)bpn;
      bv1 = *(const float4*)(bpn + 4);
      if (k0 + 2 * BK < K) {            // global_prefetch_b8 tile after that
        __builtin_prefetch(apn + BK, 0, 3);
        __builtin_prefetch(bpn + (size_t)BK * N, 0, 3);
      }
    }

#pragma unroll
    for (int kk = 0; kk < BK; kk += 4) {
      v2f a  = *(const v2f*)&As[buf][tm + mfrag][kk + kfrag];
      v2f b0 = *(const v2f*)&Bs[buf][tn +  0 + mfrag][kk + kfrag];
      v2f b1 = *(const v2f*)&Bs[buf][tn + 16 + mfrag][kk + kfrag];
      v2f b2 = *(const v2f*)&Bs[buf][tn + 32 + mfrag][kk + kfrag];
      v2f b3 = *(const v2f*)&Bs[buf][tn + 48 + mfrag][kk + kfrag];
      acc0 = __builtin_amdgcn_wmma_f32_16x16x4_f32(false, a, false, b0, (short)0, acc0, false, false);
      acc1 = __builtin_amdgcn_wmma_f32_16x16x4_f32(false, a, false, b1, (short)0, acc1, false, false);
      acc2 = __builtin_amdgcn_wmma_f32_16x16x4_f32(false, a, false, b2, (short)0, acc2, false, false);
      acc3 = __builtin_amdgcn_wmma_f32_16x16x4_f32(false, a, false, b3, (short)0, acc3, false, false);
    }

    if (more) {
      const int nb = buf ^ 1;
      *(float4*)&As[nb][arow][acol] = av;
#pragma unroll
      for (int i = 0; i < 4; ++i) {
        Bs[nb][bcol + i][brow]     = ((const float*)&bv0)[i];
        Bs[nb][bcol + 4 + i][brow] = ((const float*)&bv1)[i];
      }
      __syncthreads();
      buf = nb;
    }
  }

  // D 16x16 f32 layout: VGPR r -> M = r + 8*(lane>>4), N = lane&15
  const int ccol = bn + tn + (lane & 15);
#pragma unroll
  for (int r = 0; r < 8; ++r) {
    const int m = bm + tm + r + 8 * (lane >> 4);
    float* crow = C + (size_t)m * N + ccol;
    crow[0]  = acc0[r];
    crow[16] = acc1[r];
    crow[32] = acc2[r];
    crow[48] = acc3[r];
  }
}

// ---------------------------------------------------------------------------
// Build R7 (2 layers, 128x128) and R4 (2 layers, 16x16) from angles.
// ---------------------------------------------------------------------------
__global__ void build_rot(const float* __restrict__ ch_ang,
                          const float* __restrict__ k_ang,
                          float* __restrict__ R7, float* __restrict__ R4) {
  const int idx = blockIdx.x * blockDim.x + threadIdx.x;
  const float PI = 3.14159265358979323846f;
  if (idx < 2 * 128 * 128) {
    const int l = idx >> 14;
    const int rem = idx & 16383;
    const int i = rem >> 7, j = rem & 127;
    float p = 1.f;
#pragma unroll
    for (int q = 0; q < 7; ++q) {
      const float a = ch_ang[l * 7 + q] * PI;    // (theta*2pi)/2
      const float c = cosf(a), s = sinf(a);
      const int bi = (i >> (6 - q)) & 1;
      const int bj = (j >> (6 - q)) & 1;
      p *= (bi == bj) ? c : (bi ? s : -s);
    }
    R7[idx] = p;
  } else if (idx < 2 * 128 * 128 + 2 * 256) {
    const int r = idx - 2 * 128 * 128;
    const int l = r >> 8;
    const int rem = r & 255;
    const int i = rem >> 4, j = rem & 15;
    float p = 1.f;
#pragma unroll
    for (int q = 0; q < 4; ++q) {
      const float a = k_ang[l * 4 + q] * 0.5f;   // theta/2 (no 2pi per reference)
      const float c = cosf(a), s = sinf(a);
      const int bi = (i >> (3 - q)) & 1;
      const int bj = (j >> (3 - q)) & 1;
      p *= (bi == bj) ? c : (bi ? s : -s);
    }
    R4[r] = p;
  }
}

// out = in * kron(R7, I16):  out[i, jc*16+jk] = sum_ic in[i, ic*16+jk] * R7[ic][jc]
__global__ __launch_bounds__(256) void mul_chrot(const float* __restrict__ in,
                                                 const float* __restrict__ R7,
                                                 float* __restrict__ out) {
  const int idx = blockIdx.x * blockDim.x + threadIdx.x;   // 2048*2048
  const int i = idx >> 11;
  const int j = idx & (DIM - 1);
  const int jc = j >> 4, jk = j & 15;
  const float* row = in + (size_t)i * DIM + jk;
  float s = 0.f;
#pragma unroll 4
  for (int ic = 0; ic < 128; ++ic) s += row[ic << 4] * R7[(ic << 7) + jc];
  out[idx] = s;
}

// In-place P = P * blockdiag(I_2032, R4): only last 16 columns of each row change.
__global__ void mul_krot(float* __restrict__ P, const float* __restrict__ R4) {
  const int i = blockIdx.x * blockDim.x + threadIdx.x;   // one thread per row
  if (i >= DIM) return;
  float* row = P + (size_t)i * DIM + (DIM - 16);
  float t[16];
#pragma unroll
  for (int q = 0; q < 16; ++q) t[q] = row[q];
#pragma unroll
  for (int j = 0; j < 16; ++j) {
    float s = 0.f;
#pragma unroll
    for (int q = 0; q < 16; ++q) s += t[q] * R4[q * 16 + j];
    row[j] = s;
  }
}

// Encoding-collapse: Wt[m][i] = sum of the 4 M entries the duplicated input hits
__global__ void fold_w(const float* __restrict__ Mm, float* __restrict__ Wt) {
  const int idx = blockIdx.x * blockDim.x + threadIdx.x;   // 2048*512
  const int i = idx >> 9;
  const int m = idx & 511;
  const int c = m >> 3, d = m & 7;
  const float* row = Mm + (size_t)i * DIM + c * 32 + 2 * d;
  Wt[(size_t)m * DIM + i] = row[0] + row[1] + row[16] + row[17];
}

__global__ __launch_bounds__(256) void row_mag(const float* __restrict__ X,
                                               float* __restrict__ mag) {
  __shared__ float sm[256];
  const int b = blockIdx.x;
  const float* row = X + (size_t)b * INDIM;
  float s = 0.f;
  for (int j = threadIdx.x; j < INDIM; j += 256) { const float v = row[j]; s += v * v; }
  sm[threadIdx.x] = s;
  __syncthreads();
  for (int off = 128; off > 0; off >>= 1) {
    if (threadIdx.x < off) sm[threadIdx.x] += sm[threadIdx.x + off];
    __syncthreads();
  }
  if (threadIdx.x == 0) mag[b] = sqrtf(sm[0]) + 1e-7f;
}

// out[b] = 0.25/mag[b] * sum_i parity(i) * E[b,i]^2 , parity = +-1 via popcount
__global__ __launch_bounds__(256) void parity_reduce(const float* __restrict__ E,
                                                     const float* __restrict__ mag,
                                                     float* __restrict__ out) {
  __shared__ float sm[256];
  const int b = blockIdx.x;
  const float* row = E + (size_t)b * DIM;
  float s = 0.f;
  for (int i = threadIdx.x; i < DIM; i += 256) {
    const float v = row[i];
    s += ((__popc(i) & 1) ? -1.f : 1.f) * v * v;
  }
  sm[threadIdx.x] = s;
  __syncthreads();
  for (int off = 128; off > 0; off >>= 1) {
    if (threadIdx.x < off) sm[threadIdx.x] += sm[threadIdx.x + off];
    __syncthreads();
  }
  if (threadIdx.x == 0) out[b] = 0.25f * sm[0] / mag[b];
}

extern "C" void kernel_launch(void* const* d_in, const int* in_sizes, int n_in,
                              void* d_out, int out_size, void* d_ws, size_t ws_size,
                              hipStream_t stream) {
  const float* inputs = (const float*)d_in[0];   // (8192, 512)
  const float* ch_ang = (const float*)d_in[1];   // (2, 7)
  const float* k_ang  = (const float*)d_in[2];   // (2, 4)
  const float* c_ent  = (const float*)d_in[3];   // (2048, 2048)
  const float* k_ent  = (const float*)d_in[4];   // (2048, 2048)
  float* out = (float*)d_out;

  float* ws = (float*)d_ws;
  const size_t NM = (size_t)DIM * DIM;
  float* buf0 = ws;
  float* buf1 = buf0 + NM;
  float* E    = buf1 + NM;                       // 8192 x 2048
  float* Wt   = E + (size_t)BATCH * DIM;         // 512 x 2048
  float* R7   = Wt + (size_t)INDIM * DIM;        // 2 x 128 x 128
  float* R4   = R7 + 2 * 128 * 128;              // 2 x 16 x 16
  float* mag  = R4 + 2 * 256;                    // 8192

  build_rot<<<(2 * 128 * 128 + 2 * 256 + 255) / 256, 256, 0, stream>>>(ch_ang, k_ang, R7, R4);
  row_mag<<<BATCH, 256, 0, stream>>>(inputs, mag);

  const dim3 gM(DIM / BN, DIM / BM);
  const int nElem = (DIM * DIM) / 256;

  // M = c_ent * ch0 * c_ent * k_ent * krot0 * ch1 * c_ent * k_ent * krot1 * k_ent
  mul_chrot<<<nElem, 256, 0, stream>>>(c_ent, R7, buf0);                       // buf0 = c_ent*ch0
  gemm_nn_wmma<<<gM, 256, 0, stream>>>(buf0, c_ent, buf1, DIM, DIM, DIM);     // *c_ent
  gemm_nn_wmma<<<gM, 256, 0, stream>>>(buf1, k_ent, buf0, DIM, DIM, DIM);     // *k_ent
  mul_krot<<<(DIM + 255) / 256, 256, 0, stream>>>(buf0, R4);                  // *krot0 (in place)
  mul_chrot<<<nElem, 256, 0, stream>>>(buf0, R7 + 128 * 128, buf1);           // *ch1
  gemm_nn_wmma<<<gM, 256, 0, stream>>>(buf1, c_ent, buf0, DIM, DIM, DIM);     // *c_ent
  gemm_nn_wmma<<<gM, 256, 0, stream>>>(buf0, k_ent, buf1, DIM, DIM, DIM);     // *k_ent
  mul_krot<<<(DIM + 255) / 256, 256, 0, stream>>>(buf1, R4 + 256);            // *krot1 (in place)
  gemm_nn_wmma<<<gM, 256, 0, stream>>>(buf1, k_ent, buf0, DIM, DIM, DIM);     // *k_ent -> M

  fold_w<<<(DIM * INDIM) / 256, 256, 0, stream>>>(buf0, Wt);                  // 512 x 2048

  const dim3 gE(DIM / BN, BATCH / BM);
  gemm_nn_wmma<<<gE, 256, 0, stream>>>(inputs, Wt, E, BATCH, DIM, INDIM);     // E = X*Wt

  parity_reduce<<<BATCH, 256, 0, stream>>>(E, mag, out);
}